// QuantumDenseLayer_90426241449985
// MI455X (gfx1250) — compile-verified
//
#include <hip/hip_runtime.h>
#include <math.h>

// ---------------------------------------------------------------------------
// 12-qubit "quantum dense layer": per row, normalize -> 2 layers of
// (RY⊗12 then all-pairs CNOT cascade) -> 12 Z-expectations.
//
// RY layer factors into 3 Kronecker blocks of 16x16 (one per nibble of the
// state index); each block apply = 16 independent 16x16x16 f32 GEMMs done
// with V_WMMA_F32_16X16X4_F32 (4 accumulating K=4 steps).
// CNOT cascade = GF(2)-linear index permutation, compile-time parity masks.
// ---------------------------------------------------------------------------

typedef __attribute__((ext_vector_type(2))) float v2f;
typedef __attribute__((ext_vector_type(8))) float v8f;

#define DIM 4096

// Permutation src(k): s_after[k] = s_before[src(k)] for one full CNOT cascade
// (ctrl i, tgt j for all i<j, applied in reference order; src applies the
// involutive gate maps in reverse order). Qubit w <-> bit (11-w).
constexpr unsigned cnot_src(unsigned k) {
  for (int i = 10; i >= 0; --i)
    for (int j = 11; j >= i + 1; --j)
      k ^= ((k >> (11 - i)) & 1u) << (11 - j);
  return k;
}
// src is GF(2)-linear: bit p of src(k) = parity(k & MASK[p]).
constexpr unsigned mask_p(int p) {
  unsigned mm = 0;
  for (int b = 0; b < 12; ++b)
    mm |= ((cnot_src(1u << b) >> p) & 1u) << b;
  return mm;
}
__constant__ unsigned SRC_MASK[12] = {
    mask_p(0), mask_p(1), mask_p(2),  mask_p(3),
    mask_p(4), mask_p(5), mask_p(6),  mask_p(7),
    mask_p(8), mask_p(9), mask_p(10), mask_p(11)};

__device__ __forceinline__ unsigned src_perm(unsigned k) {
  unsigned r = 0;
#pragma unroll
  for (int p = 0; p < 12; ++p)
    r |= (unsigned)(__popc(k & SRC_MASK[p]) & 1) << p;
  return r;
}

__device__ __forceinline__ v8f wmma4(v2f a, v2f b, v8f c) {
  // D(16x16,f32) = A(16x4,f32) * B(4x16,f32) + C
  return __builtin_amdgcn_wmma_f32_16x16x4_f32(false, a, false, b,
                                               (short)0, c, false, false);
}

__global__ __launch_bounds__(256)
void qdense_kernel(const float* __restrict__ x,
                   const float* __restrict__ theta,
                   float* __restrict__ out) {
  __shared__ float st[DIM];         // state vector (unnormalized amplitudes)
  __shared__ float G[6][256];       // [layer*3+block][r*16+c] gate matrices
  __shared__ float cs[24], sn[24];  // cos/sin of theta/2
  __shared__ float wred[8];         // per-wave sum-of-squares
  __shared__ float wout[8][12];     // per-wave partial outputs

  const int t    = threadIdx.x;
  const int lane = t & 31;
  const int wv   = t >> 5;
  const int n    = lane & 15;        // A-row (M) / B-col (N) index per lane
  const int hi2  = (lane >> 4) * 2;  // K offset for lanes 16..31
  const int mrow = (lane >> 4) * 8;  // C/D row offset for lanes 16..31

  if (t < 24) {
    float s, c;
    sincosf(theta[t] * 0.5f, &s, &c);
    cs[t] = c;
    sn[t] = s;
  }
  __syncthreads();

  // Build the six 16x16 Kronecker-block gate matrices.
  // G[d*3+blk][r][c] = prod_q Ry(theta[d][blk*4+q])[bit_{3-q}(r)][bit_{3-q}(c)]
  {
    const int r = t >> 4, c = t & 15;
#pragma unroll
    for (int m = 0; m < 6; ++m) {
      float v = 1.f;
#pragma unroll
      for (int q = 0; q < 4; ++q) {
        const int rb = (r >> (3 - q)) & 1;
        const int cb = (c >> (3 - q)) & 1;
        const float cv = cs[m * 4 + q], sv = sn[m * 4 + q];
        // Ry = [[c,-s],[s,c]]
        v *= rb ? (cb ? cv : sv) : (cb ? -sv : cv);
      }
      G[m][t] = v;
    }
  }

  // Load one row of x (16 KB) coalesced b128, accumulate sum of squares.
  const float4* xr = (const float4*)(x + (size_t)blockIdx.x * DIM);
  float ss = 0.f;
#pragma unroll
  for (int e = 0; e < 4; ++e) {
    float4 v = xr[t * 4 + e];
    ((float4*)st)[t * 4 + e] = v;
    ss += v.x * v.x + v.y * v.y + v.z * v.z + v.w * v.w;
  }
#pragma unroll
  for (int o = 16; o > 0; o >>= 1) ss += __shfl_xor(ss, o, 32);
  if (lane == 0) wred[wv] = ss;
  __syncthreads();

  float sumsq = 0.f;
#pragma unroll
  for (int w = 0; w < 8; ++w) sumsq += wred[w];  // fixed order: deterministic

  const bool fb = (sumsq <= 1e-24f);  // norm <= EPS fallback -> |0...0>
  if (fb) {
#pragma unroll
    for (int e = 0; e < 16; ++e) st[e * 256 + t] = 0.f;
    if (t == 0) st[0] = 1.f;
  }
  const float inv = fb ? 1.f : 1.f / sumsq;  // circuit is norm-preserving
  __syncthreads();

#pragma unroll 1
  for (int d = 0; d < 2; ++d) {
    const float* g0 = G[d * 3 + 0];
    const float* g1 = G[d * 3 + 1];
    const float* g2 = G[d * 3 + 2];

    // ---- HI block: contract bits 11..8.  out(h,l) = sum_h' G0[h][h'] s(h',l)
    {
      v2f a[4];
#pragma unroll
      for (int kc = 0; kc < 4; ++kc) {
        a[kc].x = g0[n * 16 + kc * 4 + 0 + hi2];
        a[kc].y = g0[n * 16 + kc * 4 + 1 + hi2];
      }
#pragma unroll
      for (int tt = 0; tt < 2; ++tt) {
        const int col = (wv * 2 + tt) * 16 + n;  // this wave's exclusive cols
        v8f acc = {};
#pragma unroll
        for (int kc = 0; kc < 4; ++kc) {
          v2f b;
          b.x = st[(kc * 4 + 0 + hi2) * 256 + col];
          b.y = st[(kc * 4 + 1 + hi2) * 256 + col];
          acc = wmma4(a[kc], b, acc);
        }
#pragma unroll
        for (int v = 0; v < 8; ++v) st[(v + mrow) * 256 + col] = acc[v];
      }
    }
    __syncthreads();

    // ---- MID block: contract bits 7..4.  out(h,m,l) = sum_m' G1[m][m'] s(h,m',l)
    {
      v2f a[4];
#pragma unroll
      for (int kc = 0; kc < 4; ++kc) {
        a[kc].x = g1[n * 16 + kc * 4 + 0 + hi2];
        a[kc].y = g1[n * 16 + kc * 4 + 1 + hi2];
      }
#pragma unroll
      for (int tt = 0; tt < 2; ++tt) {
        const int base = (wv * 2 + tt) * 256;  // fixed h per tile
        v8f acc = {};
#pragma unroll
        for (int kc = 0; kc < 4; ++kc) {
          v2f b;
          b.x = st[base + (kc * 4 + 0 + hi2) * 16 + n];
          b.y = st[base + (kc * 4 + 1 + hi2) * 16 + n];
          acc = wmma4(a[kc], b, acc);
        }
#pragma unroll
        for (int v = 0; v < 8; ++v) st[base + (v + mrow) * 16 + n] = acc[v];
      }
    }
    __syncthreads();

    // ---- LO block: contract bits 3..0.  out(r,l) = sum_l' s(r,l') G2[l][l']
    //      i.e. D = S_tile * G2^T : A = S rows, B[K][N] = G2[N][K]
    {
      v2f bg[4];
#pragma unroll
      for (int kc = 0; kc < 4; ++kc) {
        bg[kc].x = g2[n * 16 + kc * 4 + 0 + hi2];
        bg[kc].y = g2[n * 16 + kc * 4 + 1 + hi2];
      }
#pragma unroll
      for (int tt = 0; tt < 2; ++tt) {
        const int rbase = (wv * 2 + tt) * 16;  // 16-row group, exclusive
        v8f acc = {};
#pragma unroll
        for (int kc = 0; kc < 4; ++kc) {
          v2f a;
          a.x = st[(rbase + n) * 16 + kc * 4 + 0 + hi2];
          a.y = st[(rbase + n) * 16 + kc * 4 + 1 + hi2];
          acc = wmma4(a, bg[kc], acc);
        }
#pragma unroll
        for (int v = 0; v < 8; ++v) st[(rbase + v + mrow) * 16 + n] = acc[v];
      }
    }
    __syncthreads();

    // CNOT cascade: materialize only after layer 0; after layer 1 it is
    // folded into the final reduction's index math.
    if (d == 0) {
      float pr[16];
#pragma unroll
      for (int e = 0; e < 16; ++e) pr[e] = st[src_perm(e * 256 + t)];
      __syncthreads();
#pragma unroll
      for (int e = 0; e < 16; ++e) st[e * 256 + t] = pr[e];
      __syncthreads();
    }
  }

  // out[w] = sum_k (1 - 2*bit_{11-w}(k)) * amp_final(k)^2 / sumsq,
  // amp_final(k) = st[src(k)] (layer-2 permutation folded in).
  float acc12[12];
#pragma unroll
  for (int w = 0; w < 12; ++w) acc12[w] = 0.f;
#pragma unroll
  for (int e = 0; e < 16; ++e) {
    const unsigned k = (unsigned)(e * 256 + t);
    const float a = st[src_perm(k)];
    const float p = a * a;
#pragma unroll
    for (int w = 0; w < 12; ++w)
      acc12[w] += ((k >> (11 - w)) & 1u) ? -p : p;
  }
#pragma unroll
  for (int w = 0; w < 12; ++w) {
#pragma unroll
    for (int o = 16; o > 0; o >>= 1)
      acc12[w] += __shfl_xor(acc12[w], o, 32);
  }
  if (lane == 0) {
#pragma unroll
    for (int w = 0; w < 12; ++w) wout[wv][w] = acc12[w];
  }
  __syncthreads();
  if (t < 12) {
    float s = 0.f;
#pragma unroll
    for (int w2 = 0; w2 < 8; ++w2) s += wout[w2][t];  // fixed order
    out[(size_t)blockIdx.x * 12 + t] = s * inv;
  }
}

extern "C" void kernel_launch(void* const* d_in, const int* in_sizes, int n_in,
                              void* d_out, int out_size, void* d_ws,
                              size_t ws_size, hipStream_t stream) {
  const float* x = (const float*)d_in[0];      // (B, 4096) f32
  const float* theta = (const float*)d_in[1];  // (2, 12) f32
  float* out = (float*)d_out;                  // (B, 12) f32
  const int B = in_sizes[0] / DIM;
  qdense_kernel<<<dim3(B), dim3(256), 0, stream>>>(x, theta, out);
}